// dsRNAPredictor_16295105921365
// MI455X (gfx1250) — compile-verified
//
#include <hip/hip_runtime.h>
#include <hip/hip_bf16.h>

typedef __attribute__((ext_vector_type(2))) float v2f;
typedef __attribute__((ext_vector_type(8))) float v8f;

#define NEG_SLOPE 0.2f
#define HC 256      // HEADS*OUT
#define COUT 64
#define ENC_NEG_INF 0x007FFFFFu

// ---- monotonic float<->uint mapping for atomic max over floats ----
__device__ __forceinline__ unsigned fenc(float f) {
  unsigned u = __float_as_uint(f);
  return (u & 0x80000000u) ? ~u : (u | 0x80000000u);
}
__device__ __forceinline__ float fdec(unsigned u) {
  return (u & 0x80000000u) ? __uint_as_float(u ^ 0x80000000u)
                           : __uint_as_float(~u);
}

// ============================================================================
// GEMM + fused attention logits:
//   H[N,256] = A[N,K] @ W[K,256]           (V_WMMA_F32_16X16X4_F32, fp32 exact)
//   a_s[n,h] = <H[n,h,:], att_src[h,:]>    (fused in epilogue; wave w == head w)
// Block = 128 threads = 4 waves; block owns 16 rows; wave w owns cols [64w,64w+64).
// A frag (16x4 f32): lanes 0-15 -> M=0..15 K={0,1}; lanes 16-31 -> K={2,3}.
// B frag (4x16 f32): lane&15 = n, lane>>4 selects K pair.
// C/D (16x16 f32): vgpr r holds row (lane>>4)*8 + r, col = lane&15.
// Full blocks (the common case: N % 16 == 0) take a guard-free path.
// ============================================================================
template <int K>
__global__ __launch_bounds__(128) void gemm_wmma_f32(
    const float* __restrict__ A, const float* __restrict__ W,
    float* __restrict__ H, const float* __restrict__ att_s,
    const float* __restrict__ att_d, float* __restrict__ as_out,
    float* __restrict__ ad_out, int N) {
  const int lane  = threadIdx.x & 31;
  const int wave  = threadIdx.x >> 5;
  const int m     = lane & 15;
  const int khalf = (lane >> 4) * 2;
  const int row0  = blockIdx.x * 16;
  const int row   = row0 + m;
  const int mb    = (lane >> 4) * 8;
  const bool full = (row0 + 16) <= N;   // block-uniform

  v2f afrag[K / 4];
  if (full) {
#pragma unroll
    for (int s = 0; s < K / 4; ++s) {
      const int k0 = s * 4 + khalf;
      afrag[s].x = A[(size_t)row * K + k0];
      afrag[s].y = A[(size_t)row * K + k0 + 1];
    }
  } else {
#pragma unroll
    for (int s = 0; s < K / 4; ++s) {
      const int k0 = s * 4 + khalf;
      afrag[s].x = (row < N) ? A[(size_t)row * K + k0] : 0.0f;
      afrag[s].y = (row < N) ? A[(size_t)row * K + k0 + 1] : 0.0f;
    }
  }

  float ps[8] = {0, 0, 0, 0, 0, 0, 0, 0};
  float pd[8] = {0, 0, 0, 0, 0, 0, 0, 0};

#pragma unroll
  for (int t = 0; t < 4; ++t) {
    const int col = wave * 64 + t * 16 + m;
    v8f c = {};
#pragma unroll
    for (int s = 0; s < K / 4; ++s) {
      const int k0 = s * 4 + khalf;
      v2f b;
      b.x = W[(size_t)k0 * HC + col];
      b.y = W[(size_t)(k0 + 1) * HC + col];
      c = __builtin_amdgcn_wmma_f32_16x16x4_f32(false, afrag[s], false, b,
                                                (short)0, c, false, false);
    }
    if (full) {
      float* hp = H + (size_t)(row0 + mb) * HC + col;
#pragma unroll
      for (int r = 0; r < 8; ++r) hp[(size_t)r * HC] = c[r];
    } else {
#pragma unroll
      for (int r = 0; r < 8; ++r) {
        const int gr = row0 + mb + r;
        if (gr < N) H[(size_t)gr * HC + col] = c[r];
      }
    }
    // fused attention-logit partials: wave w == head w, cols t*16+n
    const float aws = att_s[wave * 64 + t * 16 + m];
    const float awd = att_d[wave * 64 + t * 16 + m];
#pragma unroll
    for (int r = 0; r < 8; ++r) {
      ps[r] = fmaf(c[r], aws, ps[r]);
      pd[r] = fmaf(c[r], awd, pd[r]);
    }
  }

  // reduce over the 16 columns (stays within each half-wave) and write scores
#pragma unroll
  for (int r = 0; r < 8; ++r) {
    float s = ps[r], d = pd[r];
#pragma unroll
    for (int off = 1; off < 16; off <<= 1) {
      s += __shfl_xor(s, off);
      d += __shfl_xor(d, off);
    }
    const int gr = row0 + mb + r;
    if ((lane & 15) == 0 && gr < N) {
      as_out[(size_t)gr * 4 + wave] = s;
      ad_out[(size_t)gr * 4 + wave] = d;
    }
  }
}

// ============================================================================
// Per-layer state init: m = enc(-inf), den = 0, acc = 0
// ============================================================================
__global__ void init_softmax_state(unsigned* __restrict__ m,
                                   float* __restrict__ den,
                                   float* __restrict__ acc, int N) {
  const int stride = gridDim.x * blockDim.x;
  for (int i = blockIdx.x * blockDim.x + threadIdx.x; i < N * 4; i += stride) {
    m[i] = ENC_NEG_INF;
    den[i] = 0.0f;
  }
  for (int i = blockIdx.x * blockDim.x + threadIdx.x; i < N * COUT; i += stride)
    acc[i] = 0.0f;
}

__device__ __forceinline__ void edge_sd(const long long* __restrict__ ei, int e,
                                        int E, int& s, int& d) {
  if (e < E) {
    s = (int)ei[e];
    d = (int)ei[E + e];
  } else {
    s = e - E;  // self loop
    d = s;
  }
}

// ============================================================================
// Pass 1: segment max of leaky_relu(a_s[src] + a_d[dst]) per (dst, head)
// ============================================================================
__global__ __launch_bounds__(256) void edge_max_kernel(
    const long long* __restrict__ ei, int E, int N,
    const float* __restrict__ asrc, const float* __restrict__ adst,
    unsigned* __restrict__ m) {
  const int e = blockIdx.x * blockDim.x + threadIdx.x;
  if (e >= E + N) return;
  int s, d;
  edge_sd(ei, e, E, s, d);
  const float4 a = ((const float4*)asrc)[s];
  const float4 b = ((const float4*)adst)[d];
  float v0 = a.x + b.x, v1 = a.y + b.y, v2 = a.z + b.z, v3 = a.w + b.w;
  v0 = v0 > 0.0f ? v0 : NEG_SLOPE * v0;
  v1 = v1 > 0.0f ? v1 : NEG_SLOPE * v1;
  v2 = v2 > 0.0f ? v2 : NEG_SLOPE * v2;
  v3 = v3 > 0.0f ? v3 : NEG_SLOPE * v3;
  unsigned* mp = m + (size_t)d * 4;
  atomicMax(&mp[0], fenc(v0));
  atomicMax(&mp[1], fenc(v1));
  atomicMax(&mp[2], fenc(v2));
  atomicMax(&mp[3], fenc(v3));
}

// ============================================================================
// Pass 2: ex = exp(e - max[dst]); store ex per edge; den[dst] += ex
// ============================================================================
__global__ __launch_bounds__(256) void edge_expsum_kernel(
    const long long* __restrict__ ei, int E, int N,
    const float* __restrict__ asrc, const float* __restrict__ adst,
    const unsigned* __restrict__ m, float* __restrict__ den,
    float* __restrict__ exb) {
  const int e = blockIdx.x * blockDim.x + threadIdx.x;
  if (e >= E + N) return;
  int s, d;
  edge_sd(ei, e, E, s, d);
  const float4 a = ((const float4*)asrc)[s];
  const float4 b = ((const float4*)adst)[d];
  const uint4 mm = ((const uint4*)m)[d];
  float v0 = a.x + b.x, v1 = a.y + b.y, v2 = a.z + b.z, v3 = a.w + b.w;
  v0 = v0 > 0.0f ? v0 : NEG_SLOPE * v0;
  v1 = v1 > 0.0f ? v1 : NEG_SLOPE * v1;
  v2 = v2 > 0.0f ? v2 : NEG_SLOPE * v2;
  v3 = v3 > 0.0f ? v3 : NEG_SLOPE * v3;
  float4 ex;
  ex.x = __expf(v0 - fdec(mm.x));
  ex.y = __expf(v1 - fdec(mm.y));
  ex.z = __expf(v2 - fdec(mm.z));
  ex.w = __expf(v3 - fdec(mm.w));
  ((float4*)exb)[e] = ex;
  float* dp = den + (size_t)d * 4;
  atomicAdd(&dp[0], ex.x);
  atomicAdd(&dp[1], ex.y);
  atomicAdd(&dp[2], ex.z);
  atomicAdd(&dp[3], ex.w);
}

// ============================================================================
// Pass 3: aggregation with head-mean folded in:
//   acc[dst,c] += 0.25 * sum_h alpha[e,h] * h[src,h,c]
// One wave per edge; lane handles channels c and c+32. 64 atomics/edge (4x
// fewer than naive per-head accumulation).
// ============================================================================
__global__ __launch_bounds__(256) void edge_aggregate(
    const long long* __restrict__ ei, int E, int N,
    const float* __restrict__ H, const float* __restrict__ exb,
    const float* __restrict__ den, float* __restrict__ acc) {
  const int wid  = blockIdx.x * (blockDim.x >> 5) + (threadIdx.x >> 5);
  const int lane = threadIdx.x & 31;
  if (wid >= E + N) return;
  int s, d;
  edge_sd(ei, wid, E, s, d);
  const float* hs = H + (size_t)s * HC;
  __builtin_prefetch(hs, 0, 0);
  const float4 ex = ((const float4*)exb)[wid];
  const float4 dn = ((const float4*)den)[d];
  const float a0 = 0.25f * ex.x / dn.x;
  const float a1 = 0.25f * ex.y / dn.y;
  const float a2 = 0.25f * ex.z / dn.z;
  const float a3 = 0.25f * ex.w / dn.w;
  float r0 = a0 * hs[lane] + a1 * hs[64 + lane] + a2 * hs[128 + lane] +
             a3 * hs[192 + lane];
  float r1 = a0 * hs[32 + lane] + a1 * hs[96 + lane] + a2 * hs[160 + lane] +
             a3 * hs[224 + lane];
  float* ap = acc + (size_t)d * COUT;
  atomicAdd(&ap[lane], r0);
  atomicAdd(&ap[32 + lane], r1);
}

// out[n,c] = relu(acc[n,c] + bias[c])
__global__ void finalize_node(const float* __restrict__ acc,
                              const float* __restrict__ bias,
                              float* __restrict__ out, int N) {
  const int i = blockIdx.x * blockDim.x + threadIdx.x;
  if (i >= N * COUT) return;
  const float v = acc[i] + bias[i & 63];
  out[i] = v > 0.0f ? v : 0.0f;
}

__global__ void init_pool(float* __restrict__ pooled, float* __restrict__ cnt,
                          int G) {
  const int stride = gridDim.x * blockDim.x;
  for (int i = blockIdx.x * blockDim.x + threadIdx.x; i < G * COUT; i += stride)
    pooled[i] = 0.0f;
  for (int i = blockIdx.x * blockDim.x + threadIdx.x; i < G; i += stride)
    cnt[i] = 0.0f;
}

__global__ void pool_kernel(const float* __restrict__ x,
                            const long long* __restrict__ batch,
                            float* __restrict__ pooled, float* __restrict__ cnt,
                            int N) {
  const int i = blockIdx.x * blockDim.x + threadIdx.x;
  if (i >= N * COUT) return;
  const int n = i >> 6, c = i & 63;
  const int b = (int)batch[n];
  atomicAdd(&pooled[(size_t)b * COUT + c], x[i]);
  if (c == 0) atomicAdd(&cnt[b], 1.0f);
}

// One wave per graph: hid = relu(mean @ Wp1 + bp1) [32]; out = hid @ Wp2 + bp2
__global__ __launch_bounds__(32) void mlp_head(
    const float* __restrict__ pooled, const float* __restrict__ cnt,
    const float* __restrict__ Wp1, const float* __restrict__ bp1,
    const float* __restrict__ Wp2, const float* __restrict__ bp2,
    float* __restrict__ out, int G) {
  const int g = blockIdx.x;
  if (g >= G) return;
  const int lane = threadIdx.x;
  const float invc = 1.0f / fmaxf(cnt[g], 1.0f);
  const float* p = pooled + (size_t)g * COUT;
  float a = bp1[lane];
#pragma unroll
  for (int i = 0; i < COUT; ++i) a = fmaf(p[i] * invc, Wp1[i * 32 + lane], a);
  float contrib = fmaxf(a, 0.0f) * Wp2[lane];
#pragma unroll
  for (int off = 16; off; off >>= 1) contrib += __shfl_xor(contrib, off);
  if (lane == 0) out[g] = contrib + bp2[0];
}

// ============================================================================
extern "C" void kernel_launch(void* const* d_in, const int* in_sizes, int n_in,
                              void* d_out, int out_size, void* d_ws,
                              size_t ws_size, hipStream_t stream) {
  (void)n_in;
  (void)ws_size;
  const float*     x     = (const float*)d_in[0];
  const long long* ei    = (const long long*)d_in[1];
  const long long* batch = (const long long*)d_in[2];
  const float* W1  = (const float*)d_in[4];
  const float* as1 = (const float*)d_in[5];
  const float* ad1 = (const float*)d_in[6];
  const float* b1  = (const float*)d_in[7];
  const float* W2  = (const float*)d_in[8];
  const float* as2 = (const float*)d_in[9];
  const float* ad2 = (const float*)d_in[10];
  const float* b2  = (const float*)d_in[11];
  const float* Wp1 = (const float*)d_in[12];
  const float* bp1 = (const float*)d_in[13];
  const float* Wp2 = (const float*)d_in[14];
  const float* bp2 = (const float*)d_in[15];

  const int N = in_sizes[0] / 8;   // Fin = 8
  const int E = in_sizes[1] / 2;   // edge_index [2,E]
  const int G = out_size;          // output is [G,1]
  const int Etot = E + N;          // with self-loops

  // --- carve workspace (256B aligned slices) ---
  char* base = (char*)d_ws;
  size_t off = 0;
  auto carve = [&](size_t bytes) -> char* {
    char* p = base + off;
    off += (bytes + 255) & ~(size_t)255;
    return p;
  };
  float*    h_buf  = (float*)carve((size_t)N * HC * 4);    // 102.4 MB
  float*    accb   = (float*)carve((size_t)N * COUT * 4);  // 25.6 MB
  float*    nodeo  = (float*)carve((size_t)N * COUT * 4);  // 25.6 MB
  float*    a_s    = (float*)carve((size_t)N * 4 * 4);
  float*    a_d    = (float*)carve((size_t)N * 4 * 4);
  unsigned* mbuf   = (unsigned*)carve((size_t)N * 4 * 4);
  float*    den    = (float*)carve((size_t)N * 4 * 4);
  float*    exb    = (float*)carve((size_t)Etot * 4 * 4);  // 14.4 MB
  float*    pooled = (float*)carve((size_t)G * COUT * 4);
  float*    cnt    = (float*)carve((size_t)G * 4);

  const int gEdge  = (Etot + 255) / 256;
  const int gEdgeW = (Etot + 7) / 8;        // wave-per-edge, 8 waves/block
  const int gNode  = (N * COUT + 255) / 256;
  const int gGemm  = (N + 15) / 16;

  // ---------------- Layer 1 ----------------
  gemm_wmma_f32<8><<<gGemm, 128, 0, stream>>>(x, W1, h_buf, as1, ad1, a_s,
                                              a_d, N);
  init_softmax_state<<<2048, 256, 0, stream>>>(mbuf, den, accb, N);
  edge_max_kernel<<<gEdge, 256, 0, stream>>>(ei, E, N, a_s, a_d, mbuf);
  edge_expsum_kernel<<<gEdge, 256, 0, stream>>>(ei, E, N, a_s, a_d, mbuf, den,
                                                exb);
  edge_aggregate<<<gEdgeW, 256, 0, stream>>>(ei, E, N, h_buf, exb, den, accb);
  finalize_node<<<gNode, 256, 0, stream>>>(accb, b1, nodeo, N);

  // ---------------- Layer 2 ----------------
  gemm_wmma_f32<64><<<gGemm, 128, 0, stream>>>(nodeo, W2, h_buf, as2, ad2, a_s,
                                               a_d, N);
  init_softmax_state<<<2048, 256, 0, stream>>>(mbuf, den, accb, N);
  edge_max_kernel<<<gEdge, 256, 0, stream>>>(ei, E, N, a_s, a_d, mbuf);
  edge_expsum_kernel<<<gEdge, 256, 0, stream>>>(ei, E, N, a_s, a_d, mbuf, den,
                                                exb);
  edge_aggregate<<<gEdgeW, 256, 0, stream>>>(ei, E, N, h_buf, exb, den, accb);
  finalize_node<<<gNode, 256, 0, stream>>>(accb, b2, nodeo, N);

  // ---------------- Pool + MLP head ----------------
  init_pool<<<64, 256, 0, stream>>>(pooled, cnt, G);
  pool_kernel<<<gNode, 256, 0, stream>>>(nodeo, batch, pooled, cnt, N);
  mlp_head<<<G, 32, 0, stream>>>(pooled, cnt, Wp1, bp1, Wp2, bp2,
                                 (float*)d_out, G);
}